// AttackFortifyNet_32744830665071
// MI455X (gfx1250) — compile-verified
//
#include <hip/hip_runtime.h>
#include <hip/hip_bf16.h>
#include <math.h>

#define T_SZ   8192
#define H_SZ   4096
#define NACT   65536
#define NWORDS (T_SZ * (T_SZ / 32))   /* 2,097,152 bitmask words == 8 MB */
#define NPART  1024

typedef float v2f __attribute__((ext_vector_type(2)));
typedef float v8f __attribute__((ext_vector_type(8)));

// ---- monotone float<->u32 encoding so atomicMax(u32) == float max ----
__device__ __forceinline__ unsigned enc_f32(float f) {
  unsigned u = __float_as_uint(f);
  return (u & 0x80000000u) ? ~u : (u | 0x80000000u);
}
__device__ __forceinline__ float dec_f32(unsigned e) {
  unsigned u = (e & 0x80000000u) ? (e & 0x7fffffffu) : ~e;
  return __uint_as_float(u);
}

__device__ __forceinline__ float block_reduce_sum(float v) {
  __shared__ float sh[8];
  #pragma unroll
  for (int off = 16; off > 0; off >>= 1) v += __shfl_xor(v, off, 32);
  const int wid = threadIdx.x >> 5;
  if ((threadIdx.x & 31) == 0) sh[wid] = v;
  __syncthreads();
  const int nw = blockDim.x >> 5;
  v = (threadIdx.x < nw) ? sh[threadIdx.x] : 0.0f;
  if (wid == 0) {
    #pragma unroll
    for (int off = 4; off > 0; off >>= 1) v += __shfl_xor(v, off, 32);
  }
  return v; // valid on thread 0
}

// -------- init: clear bitmask + scalars (ws is poisoned 0xAA by harness) ----
__global__ void init_ws_kernel(unsigned* __restrict__ bits,
                               unsigned* __restrict__ maxenc,
                               float* __restrict__ partials) {
  const int stride = gridDim.x * blockDim.x;
  int t = blockIdx.x * blockDim.x + threadIdx.x;
  for (int i = t; i < NWORDS; i += stride) bits[i] = 0u;
  if (t < NPART) partials[t] = 0.0f;
  if (t == 0) *maxenc = 0u;   // 0 is the minimum of the encoded order
}

// -------- GEMV y = tanh(W @ v + b) via V_WMMA_F32_16X16X4_F32 --------------
// blockDim.x == 32 (one full wave, EXEC all ones). Each wave owns 16 rows.
// A fragment (16x4 f32 ISA layout): lane<16 -> (M=lane, K={0,1}),
//                                   lane>=16 -> (M=lane-16, K={2,3}).
// B fragment broadcasts v[k..k+3] across all 16 columns -> every column of C
// holds the same dot product.
__global__ void gemv_tanh_wmma(const float* __restrict__ W,
                               const float* __restrict__ vec,
                               const float* __restrict__ bias,
                               float* __restrict__ out,
                               int K) {
  const int lane = threadIdx.x;
  const int row0 = blockIdx.x * 16;
  const int sub  = lane & 15;
  const int hi   = lane >> 4;                       // 0 or 1
  const float* __restrict__ wrow = W + (size_t)(row0 + sub) * (size_t)K + 2 * hi;
  const float* __restrict__ vrow = vec + 2 * hi;

  v8f c = {0.f, 0.f, 0.f, 0.f, 0.f, 0.f, 0.f, 0.f};
  for (int k = 0; k < K; k += 32) {
    __builtin_prefetch(wrow + k + 512, 0, 0);       // global_prefetch_b8, ~2KB ahead
    #pragma unroll
    for (int kk = 0; kk < 32; kk += 4) {
      v2f a, b;
      a.x = wrow[k + kk];
      a.y = wrow[k + kk + 1];
      b.x = vrow[k + kk];
      b.y = vrow[k + kk + 1];
      c = __builtin_amdgcn_wmma_f32_16x16x4_f32(false, a, false, b,
                                                (short)0, c, false, false);
    }
  }

  // All N columns of C are identical: row m lives in c[m&7],
  // lanes 0..15 for m<8, lanes 16..31 for m>=8. Route row m to lane m.
  float acc = 0.0f;
  #pragma unroll
  for (int m = 0; m < 16; ++m) {
    float v = __shfl(c[m & 7], (m < 8) ? 0 : 16, 32);
    if (lane == m) acc = v;
  }
  if (lane < 16) {
    const int r = row0 + lane;
    out[r] = tanhf(acc + bias[r]);
  }
}

// -------- set bitmask + global max over action values ----------------------
__global__ void scatter_max_kernel(const int* __restrict__ pa,
                                   const float* __restrict__ torig,
                                   const float* __restrict__ tdest,
                                   unsigned* __restrict__ bits,
                                   unsigned* __restrict__ maxenc) {
  const int i = blockIdx.x * blockDim.x + threadIdx.x;
  float v = -3.0e38f;
  if (i < NACT) {
    const int o = pa[2 * i];
    const int d = pa[2 * i + 1];
    const unsigned pos = (unsigned)o * (unsigned)T_SZ + (unsigned)d;
    atomicOr(&bits[pos >> 5], 1u << (pos & 31u));
    v = torig[o] * tdest[d];
  }
  #pragma unroll
  for (int off = 16; off > 0; off >>= 1) v = fmaxf(v, __shfl_xor(v, off, 32));
  if ((threadIdx.x & 31) == 0) atomicMax(maxenc, enc_f32(v));
}

// -------- deterministic exp-sum over set bits (fixed reduction tree) -------
__global__ void sumbits_kernel(const unsigned* __restrict__ bits,
                               const float* __restrict__ torig,
                               const float* __restrict__ tdest,
                               const unsigned* __restrict__ maxenc,
                               float* __restrict__ partials) {
  const float mx = dec_f32(*maxenc);
  float s = 0.0f;
  const int stride = gridDim.x * blockDim.x;
  for (int w = blockIdx.x * blockDim.x + threadIdx.x; w < NWORDS; w += stride) {
    unsigned b = bits[w];
    if (b) {
      const int base = w << 5;                 // 32 positions share one row o
      const float to = torig[base >> 13];      // T = 2^13
      const float* __restrict__ td = tdest + (base & (T_SZ - 1));
      while (b) {
        const int j = __ffs(b) - 1;
        b &= b - 1;
        s += expf(to * td[j] - mx);
      }
    }
  }
  s = block_reduce_sum(s);
  if (threadIdx.x == 0) partials[blockIdx.x] = s;
}

__global__ void reduce_kernel(const float* __restrict__ partials,
                              float* __restrict__ inv_sum) {
  float s = 0.0f;
  for (int i = threadIdx.x; i < NPART; i += blockDim.x) s += partials[i];
  s = block_reduce_sum(s);
  if (threadIdx.x == 0) *inv_sum = 1.0f / s;
}

// -------- single streaming pass writing all 64M outputs --------------------
// One thread per 32-element (128B) chunk; masked entries are exactly 0.0f.
__global__ void out_kernel(const unsigned* __restrict__ bits,
                           const float* __restrict__ torig,
                           const float* __restrict__ tdest,
                           const unsigned* __restrict__ maxenc,
                           const float* __restrict__ inv_sum_p,
                           float* __restrict__ out) {
  const float mx   = dec_f32(*maxenc);
  const float isum = *inv_sum_p;
  const int stride = gridDim.x * blockDim.x;
  for (int w = blockIdx.x * blockDim.x + threadIdx.x; w < NWORDS; w += stride) {
    const unsigned b = bits[w];
    float4* __restrict__ dst = (float4*)(out + ((size_t)w << 5));
    if (!b) {
      const float4 z = {0.f, 0.f, 0.f, 0.f};
      #pragma unroll
      for (int q = 0; q < 8; ++q) dst[q] = z;
    } else {
      const int base = w << 5;
      const float to = torig[base >> 13];
      const float* __restrict__ td = tdest + (base & (T_SZ - 1));
      #pragma unroll
      for (int q = 0; q < 8; ++q) {
        float4 r;
        r.x = ((b >> (4 * q + 0)) & 1u) ? expf(to * td[4 * q + 0] - mx) * isum : 0.0f;
        r.y = ((b >> (4 * q + 1)) & 1u) ? expf(to * td[4 * q + 1] - mx) * isum : 0.0f;
        r.z = ((b >> (4 * q + 2)) & 1u) ? expf(to * td[4 * q + 2] - mx) * isum : 0.0f;
        r.w = ((b >> (4 * q + 3)) & 1u) ? expf(to * td[4 * q + 3] - mx) * isum : 0.0f;
        dst[q] = r;
      }
    }
  }
}

extern "C" void kernel_launch(void* const* d_in, const int* in_sizes, int n_in,
                              void* d_out, int out_size, void* d_ws, size_t ws_size,
                              hipStream_t stream) {
  const int*   pa       = (const int*)  d_in[0];  // [NACT, 2] int32
  const float* presence = (const float*)d_in[1];  // [T]
  const float* W_in     = (const float*)d_in[2];  // [H, T]
  const float* b_in     = (const float*)d_in[3];  // [H]
  const float* W_orig   = (const float*)d_in[4];  // [T, H]
  const float* b_orig   = (const float*)d_in[5];  // [T]
  const float* W_dest   = (const float*)d_in[6];  // [T, H]
  const float* b_dest   = (const float*)d_in[7];  // [T]
  float* out = (float*)d_out;

  // workspace layout: 8MB bitmask | x[H] | torig[T] | tdest[T] | partials | scalars
  unsigned* bits     = (unsigned*)d_ws;
  float*    x        = (float*)((char*)d_ws + (size_t)NWORDS * 4);
  float*    torig    = x + H_SZ;
  float*    tdest    = torig + T_SZ;
  float*    partials = tdest + T_SZ;
  unsigned* maxenc   = (unsigned*)(partials + NPART);
  float*    inv_sum  = (float*)(maxenc + 1);

  init_ws_kernel<<<2048, 256, 0, stream>>>(bits, maxenc, partials);

  gemv_tanh_wmma<<<H_SZ / 16, 32, 0, stream>>>(W_in,   presence, b_in,   x,     T_SZ);
  gemv_tanh_wmma<<<T_SZ / 16, 32, 0, stream>>>(W_orig, x,        b_orig, torig, H_SZ);
  gemv_tanh_wmma<<<T_SZ / 16, 32, 0, stream>>>(W_dest, x,        b_dest, tdest, H_SZ);

  scatter_max_kernel<<<NACT / 256, 256, 0, stream>>>(pa, torig, tdest, bits, maxenc);
  sumbits_kernel<<<NPART, 256, 0, stream>>>(bits, torig, tdest, maxenc, partials);
  reduce_kernel<<<1, 256, 0, stream>>>(partials, inv_sum);
  out_kernel<<<4096, 256, 0, stream>>>(bits, torig, tdest, maxenc, inv_sum, out);
}